// FeatureEmbedder_2791728742828
// MI455X (gfx1250) — compile-verified
//
#include <hip/hip_runtime.h>

typedef __attribute__((ext_vector_type(2))) float v2f;
typedef __attribute__((ext_vector_type(8))) float v8f;

#define BATCH_N    16384
#define NFEAT      128
#define EMBD       64
#define OUT_STRIDE (NFEAT * EMBD)   // 8192 floats per output row

// One wave (32 lanes) handles a 16-row x 1-feature stripe:
//   out[r, f*64 + c] = relu(x[r,f] * W[f,c] + b[f,c]),  r in [row_base, row_base+16)
// computed as four f32 WMMA 16x16 outer-product tiles (K=4, only K=0 nonzero):
//   A[m][0] = x[row_base+m, f]   (A lanes 0-15, VGPR0 = K=0 per ISA layout)
//   B[k][n] = W[f, col0+n]       (replicated into every B slot; only K=0 used)
//   C[m][n] = b[f, col0+n]       (bias enters through the accumulator)
__global__ __launch_bounds__(256) void feature_embed_wmma(
    const float* __restrict__ x, const float* __restrict__ W,
    const float* __restrict__ bias, float* __restrict__ out) {
  const int lane    = threadIdx.x & 31;
  const int wave    = threadIdx.x >> 5;
  const int tile    = blockIdx.x * 8 + wave;   // 131072 tiles total
  const int f       = tile & (NFEAT - 1);
  const int rowBase = (tile >> 7) << 4;        // (tile / 128) * 16

  const int  laneLo = lane & 15;
  const bool hiHalf = lane >= 16;

  // A-matrix (16x4 f32): x values in K=0 column only.
  const float xv = x[(rowBase + laneLo) * NFEAT + f];
  v2f A;
  A.x = hiHalf ? 0.0f : xv;  // VGPR0: lanes 0-15 hold K=0 (M=lane)
  A.y = 0.0f;                // K=1 / K=3 columns are zero

  // D layout: VGPR r, lanes 0-15 -> (M=r, N=lane); lanes 16-31 -> (M=8+r).
  float* outp = out + (size_t)(rowBase + (hiHalf ? 8 : 0)) * OUT_STRIDE
                    + f * EMBD + laneLo;

  const float* Wf = W    + f * EMBD;
  const float* bf = bias + f * EMBD;

#pragma unroll
  for (int ct = 0; ct < 4; ++ct) {
    const int col = ct * 16 + laneLo;
    const float wv = Wf[col];
    const float bv = bf[col];

    // B (4x16 f32): every (K,N) slot = W[f, col0+N]; only K=0 contributes
    // since A's K!=0 entries are zero, so B's exact K-row placement is moot.
    v2f B;
    B.x = wv;
    B.y = wv;

    // C (16x16 f32): bias broadcast down the M dimension.
    v8f C = {bv, bv, bv, bv, bv, bv, bv, bv};

    v8f D = __builtin_amdgcn_wmma_f32_16x16x4_f32(
        /*neg_a=*/false, A, /*neg_b=*/false, B,
        /*c_mod=*/(short)0, C, /*reuse_a=*/false, /*reuse_b=*/false);

#pragma unroll
    for (int r = 0; r < 8; ++r) {
      // lanes 0-15 write 64 contiguous bytes of row (rowBase+r),
      // lanes 16-31 the matching 64B of row (rowBase+8+r).
      outp[(size_t)r * OUT_STRIDE + ct * 16] = fmaxf(D[r], 0.0f);
    }
  }
}

extern "C" void kernel_launch(void* const* d_in, const int* in_sizes, int n_in,
                              void* d_out, int out_size, void* d_ws, size_t ws_size,
                              hipStream_t stream) {
  (void)in_sizes; (void)n_in; (void)out_size; (void)d_ws; (void)ws_size;
  const float* x    = (const float*)d_in[0];  // [16384, 128]
  const float* W    = (const float*)d_in[1];  // [128, 64]
  const float* bias = (const float*)d_in[2];  // [128, 64]
  float* out        = (float*)d_out;          // [16384, 8192]

  // 131072 waves (one per 16-row x 1-feature stripe) = 16384 blocks x 8 waves.
  const int tiles  = (BATCH_N / 16) * NFEAT;  // 131072
  const int blocks = tiles / 8;               // 16384
  feature_embed_wmma<<<dim3(blocks), dim3(256), 0, stream>>>(x, W, bias, out);
}